// NTXentLossManhattan_33775622816381
// MI455X (gfx1250) — compile-verified
//
#include <hip/hip_runtime.h>
#include <math.h>

#define N_ROWS   1024
#define DDIM     256
#define TILE_I   16
#define CHUNK_J  32
#define JPART    256                    // j-range per block (grid.y = 4)
#define CHUNKS_PER_PART (JPART / CHUNK_J)

typedef __attribute__((ext_vector_type(2))) float        v2f;
typedef __attribute__((ext_vector_type(8))) float        v8f;
typedef __attribute__((ext_vector_type(4))) unsigned int v4u;
typedef __attribute__((ext_vector_type(4))) int          v4i;
typedef __attribute__((ext_vector_type(8))) int          v8i;

// ---------------------------------------------------------------------------
// TDM: async DMA of `nrows` x 256 f32 rows from global into LDS (2-D tile).
// D# built per CDNA5 ISA (group0 128b, group1 256b,
// groups 2/3 zero for a 2-D tensor). Tracked by TENSORcnt.
// amdgpu-toolchain (clang-23) builtin arity: 6 args
//   (uint32x4 g0, int32x8 g1, int32x4 g2, int32x4 g3, int32x8 pad, i32 cpol)
// ---------------------------------------------------------------------------
__device__ inline void tdm_load_rows_to_lds(const float* gsrc,
                                            unsigned lds_byte_addr, int nrows) {
  unsigned long long ga = (unsigned long long)(uintptr_t)gsrc;
  v4u g0;
  g0[0] = 1u;                                            // count=1, user mode
  g0[1] = lds_byte_addr;                                 // lds_addr
  g0[2] = (unsigned)(ga & 0xffffffffu);                  // global_addr[31:0]
  g0[3] = (unsigned)((ga >> 32) & 0x1ffffffu)            // global_addr[56:32]
        | (2u << 30);                                    // type = 2 ("image")
  v8i g1;
  g1[0] = (int)(2u << 16);                               // data_size = 4B
  g1[1] = (int)((unsigned)DDIM << 16);                   // tensor_dim0 = 256
  g1[2] = (int)(((unsigned)N_ROWS & 0xffffu) << 16);     // tensor_dim1 = 1024
  g1[3] = (int)((unsigned)DDIM << 16);                   // tile_dim0 = 256
  g1[4] = (int)(nrows & 0xffff);                         // tile_dim1 = nrows
  g1[5] = (int)DDIM;                                     // tensor_dim0_stride
  g1[6] = 0;
  g1[7] = 0;
  v4i z4 = {0, 0, 0, 0};
  v8i z8 = {0, 0, 0, 0, 0, 0, 0, 0};
  __builtin_amdgcn_tensor_load_to_lds(g0, g1, z4, z4, z8, 0);
}

// ---------------------------------------------------------------------------
// WMMA Gram-diagonal L2 norms for a 16x256 LDS tile. G = Z * Z^T via 64
// chained V_WMMA_F32_16X16X4_F32; diag(G)[m] = ||row m||^2. Must be called by
// one full wave (EXEC all ones). Writes 1/max(||row||, eps) to inv_out[0..15].
// A 16x4 layout: lanes 0-15 hold K={0,1}, lanes 16-31 hold K={2,3} of row
// M=lane%16; B 4x16 of Z^T has the identical per-lane values, so frag B == A.
// ---------------------------------------------------------------------------
__device__ inline void gram_inv_norms_16(const float* tile, float* inv_out) {
  const int lane = threadIdx.x & 31;
  const int row  = lane & 15;
  const int koff = (lane >> 4) << 1;          // 0 or 2
  v8f acc = {};
  const float* p = tile + row * DDIM + koff;
#pragma unroll 4
  for (int k0 = 0; k0 < DDIM; k0 += 4) {
    v2f a;
    a.x = p[k0];
    a.y = p[k0 + 1];
    acc = __builtin_amdgcn_wmma_f32_16x16x4_f32(
        false, a, false, a, (short)0, acc, false, false);
  }
  // Diagonal of 16x16 f32 C: element (m,m) lives in VGPR m%8, lane m (m<8)
  // or lane m+16 (m>=8).
  if (lane < 8 || lane >= 24) {
    const int sel = (lane < 8) ? lane : (lane - 24);
    float d = acc[0];
#pragma unroll
    for (int r = 1; r < 8; ++r)
      if (sel == r) d = acc[r];
    const int m = (lane < 8) ? lane : (lane - 16);
    inv_out[m] = 1.0f / fmaxf(sqrtf(d), 1e-12f);
  }
}

// ---------------------------------------------------------------------------
// Main kernel: block (bi,bj) handles i-tile bi (16 rows) x j-range bj (256 j).
// Emits per-row partial logsumexp state (M,S) and the diagonal term.
// ---------------------------------------------------------------------------
__global__ __launch_bounds__(256) void ntxent_man_partial(
    const float* __restrict__ zis, const float* __restrict__ zjs,
    float* __restrict__ partM, float* __restrict__ partS,
    float* __restrict__ posArr) {
  __shared__ __align__(16) float ldsA[TILE_I * DDIM];   // 16 KB
  __shared__ __align__(16) float ldsB[CHUNK_J * DDIM];  // 32 KB
  __shared__ float invI[TILE_I];
  __shared__ float invJ[CHUNK_J];
  __shared__ float pmS[256], psS[256];
  __shared__ float posS[TILE_I];

  const int t       = threadIdx.x;
  const int bi      = blockIdx.x;
  const int bj      = blockIdx.y;
  const int i       = t & 15;
  const int q       = t >> 4;
  const int iglobal = bi * TILE_I + i;
  const int wave    = t >> 5;
  const unsigned ldsB_addr = (unsigned)(unsigned long long)(uintptr_t)(void*)&ldsB[0];

  // ---- stage + normalize the zi tile (norms via the matrix pipe) ----
  {
    const float4* src  = (const float4*)(zis + (size_t)bi * TILE_I * DDIM);
    float4*       dstA = (float4*)ldsA;
    for (int idx = t; idx < TILE_I * DDIM / 4; idx += 256) dstA[idx] = src[idx];
  }
  if (t < TILE_I) posS[t] = 0.0f;
  __syncthreads();
  if (wave == 0) gram_inv_norms_16(ldsA, invI);
  __syncthreads();
  for (int idx = t; idx < TILE_I * DDIM; idx += 256) ldsA[idx] *= invI[idx >> 8];

  float mmax = -INFINITY, ssum = 0.0f;
  __syncthreads();

  for (int c = 0; c < CHUNKS_PER_PART; ++c) {
    const int j0 = bj * JPART + c * CHUNK_J;

    // ---- async-stage the zj chunk via the Tensor Data Mover ----
    if (wave == 0) {
      tdm_load_rows_to_lds(zjs + (size_t)j0 * DDIM, ldsB_addr, CHUNK_J);
      __builtin_amdgcn_s_wait_tensorcnt(0);
    }
    __syncthreads();

    // ---- zj chunk norms: two 16-row Grams on the matrix pipe ----
    if (wave == 0) {
      gram_inv_norms_16(ldsB, invJ);
      gram_inv_norms_16(ldsB + 16 * DDIM, invJ + 16);
    }
    __syncthreads();
    for (int idx = t; idx < CHUNK_J * DDIM; idx += 256) ldsB[idx] *= invJ[idx >> 8];
    __syncthreads();

    // ---- main VALU loop: thread owns (i, j0+q) and (i, j0+q+16) ----
    const float4* arow = (const float4*)(ldsA + i * DDIM);
    const float4* b0r  = (const float4*)(ldsB + q * DDIM);
    const float4* b1r  = (const float4*)(ldsB + (q + 16) * DDIM);
    float m0 = 0.0f, m1 = 0.0f;
#pragma unroll 8
    for (int d = 0; d < DDIM / 4; ++d) {
      const float4 a  = arow[d];
      const float4 b0 = b0r[d];
      const float4 b1 = b1r[d];
      m0 += fabsf(a.x - b0.x) + fabsf(a.y - b0.y) +
            fabsf(a.z - b0.z) + fabsf(a.w - b0.w);
      m1 += fabsf(a.x - b1.x) + fabsf(a.y - b1.y) +
            fabsf(a.z - b1.z) + fabsf(a.w - b1.w);
    }
    const float v0 = -2.0f * m0;  // sim = -man / TEMPERATURE, T = 0.5
    const float v1 = -2.0f * m1;

    if (j0 + q      == iglobal) posS[i] = v0;
    if (j0 + q + 16 == iglobal) posS[i] = v1;

    // online logsumexp (per thread, over its 2 j's per chunk)
    if (v0 > mmax) { ssum = ssum * __expf(mmax - v0) + 1.0f; mmax = v0; }
    else           { ssum += __expf(v0 - mmax); }
    if (v1 > mmax) { ssum = ssum * __expf(mmax - v1) + 1.0f; mmax = v1; }
    else           { ssum += __expf(v1 - mmax); }

    __syncthreads();  // protect ldsB before next chunk's TDM overwrite
  }

  // ---- combine the 16 per-thread partials of each i-row ----
  pmS[t] = mmax;
  psS[t] = ssum;
  __syncthreads();
  if (t < TILE_I) {
    float M = -INFINITY;
#pragma unroll
    for (int qq = 0; qq < 16; ++qq) M = fmaxf(M, pmS[t + 16 * qq]);
    float S = 0.0f;
#pragma unroll
    for (int qq = 0; qq < 16; ++qq)
      S += psS[t + 16 * qq] * __expf(pmS[t + 16 * qq] - M);
    const int r = bi * TILE_I + t;
    partM[r * 4 + bj] = M;
    partS[r * 4 + bj] = S;
    if ((r >> 8) == bj) posArr[r] = posS[t];  // this j-quarter held the diagonal
  }
}

// ---------------------------------------------------------------------------
// Final: merge the 4 j-quarter LSE partials per row, subtract pos, mean.
// Single block => deterministic (no float atomics).
// ---------------------------------------------------------------------------
__global__ __launch_bounds__(256) void ntxent_final(
    const float* __restrict__ partM, const float* __restrict__ partS,
    const float* __restrict__ posArr, float* __restrict__ out) {
  __shared__ float red[256];
  const int t = threadIdx.x;
  float local = 0.0f;
  for (int r = t; r < N_ROWS; r += 256) {
    float M = partM[r * 4];
#pragma unroll
    for (int p = 1; p < 4; ++p) M = fmaxf(M, partM[r * 4 + p]);
    float S = 0.0f;
#pragma unroll
    for (int p = 0; p < 4; ++p)
      S += partS[r * 4 + p] * __expf(partM[r * 4 + p] - M);
    local += (M + __logf(S)) - posArr[r];
  }
  red[t] = local;
  __syncthreads();
  for (int s = 128; s > 0; s >>= 1) {
    if (t < s) red[t] += red[t + s];
    __syncthreads();
  }
  if (t == 0) out[0] = red[0] * (1.0f / N_ROWS);
}

extern "C" void kernel_launch(void* const* d_in, const int* in_sizes, int n_in,
                              void* d_out, int out_size, void* d_ws, size_t ws_size,
                              hipStream_t stream) {
  const float* zis = (const float*)d_in[0];
  const float* zjs = (const float*)d_in[1];

  float* partM  = (float*)d_ws;            // 1024*4 floats
  float* partS  = partM + N_ROWS * 4;      // 1024*4 floats
  float* posArr = partS + N_ROWS * 4;      // 1024 floats   (36 KB total)

  dim3 grid(N_ROWS / TILE_I, N_ROWS / JPART);  // 64 x 4
  ntxent_man_partial<<<grid, 256, 0, stream>>>(zis, zjs, partM, partS, posArr);
  ntxent_final<<<1, 256, 0, stream>>>(partM, partS, posArr, (float*)d_out);
}